// ConvCheb_40355512713613
// MI455X (gfx1250) — compile-verified
//
#include <hip/hip_runtime.h>

// ---------------- problem constants ----------------
constexpr int kB    = 8;
constexpr int kV    = 100000;
constexpr int kFin  = 32;
constexpr int kK    = 4;
constexpr int kFout = 64;
constexpr int kE    = 800000;
constexpr int kC    = kB * kFin;   // 256 features per vertex row
constexpr int kKK   = kK * kFin;   // 128 contraction dims
constexpr int LDA   = 132;         // padded LDS strides (conflict-free 2h+j b64 reads)
constexpr int LDW   = 132;

typedef __attribute__((ext_vector_type(2))) float v2f;
typedef __attribute__((ext_vector_type(8))) float v8f;

// x0[v*256 + b*32 + f] = inputs[b][v][f]
__global__ __launch_bounds__(256) void transpose_in(const float* __restrict__ in,
                                                    float* __restrict__ x0) {
    size_t tid = (size_t)blockIdx.x * 256 + threadIdx.x;   // 25.6M
    int c = (int)(tid & 255);
    size_t v = tid >> 8;
    int b = c >> 5, f = c & 31;
    x0[tid] = in[(size_t)b * kV * kFin + v * kFin + f];
}

// ---------------- CSR construction (per call; stateless) ----------------
__global__ __launch_bounds__(256) void zero_ints(int* __restrict__ p, int n) {
    int i = blockIdx.x * 256 + threadIdx.x;
    if (i < n) p[i] = 0;
}

__global__ __launch_bounds__(256) void count_deg(const int* __restrict__ rows,
                                                 int* __restrict__ deg) {
    int e = blockIdx.x * 256 + threadIdx.x;
    if (e < kE) atomicAdd(&deg[rows[e]], 1);
}

// single-workgroup exclusive scan: rowstart[v] = sum(deg[0..v-1]), rowstart[V] = E
__global__ __launch_bounds__(1024) void scan_deg(const int* __restrict__ deg,
                                                 int* __restrict__ rowstart) {
    __shared__ int partial[1024];
    const int t = threadIdx.x;
    const int per = (kV + 1023) / 1024;    // 98
    const int base = t * per;
    int s = 0;
    for (int i = 0; i < per; ++i) {
        int idx = base + i;
        if (idx < kV) s += deg[idx];
    }
    partial[t] = s;
    __syncthreads();
    for (int off = 1; off < 1024; off <<= 1) {
        int v = 0;
        if (t >= off) v = partial[t - off];
        __syncthreads();
        if (t >= off) partial[t] += v;
        __syncthreads();
    }
    int run = (t == 0) ? 0 : partial[t - 1];
    for (int i = 0; i < per; ++i) {
        int idx = base + i;
        if (idx < kV) {
            rowstart[idx] = run;
            run += deg[idx];
        }
    }
    if (t == 0) rowstart[kV] = kE;
}

__global__ __launch_bounds__(256) void copy_ints(const int* __restrict__ src,
                                                 int* __restrict__ dst, int n) {
    int i = blockIdx.x * 256 + threadIdx.x;
    if (i < n) dst[i] = src[i];
}

__global__ __launch_bounds__(256) void fill_csr(const int* __restrict__ rows,
                                                const int* __restrict__ cols,
                                                const float* __restrict__ vals,
                                                int* __restrict__ pos,
                                                int* __restrict__ ecol,
                                                float* __restrict__ eval) {
    int e = blockIdx.x * 256 + threadIdx.x;
    if (e >= kE) return;
    int idx = atomicAdd(&pos[rows[e]], 1);
    ecol[idx] = cols[e];
    eval[idx] = vals[e];
}

// ---------------- gather-only SpMM: y[row] = alpha*sum(val*x[col]) - xprev[row]
// 64 threads per row, float4 feature slice each; x rows are 1KB contiguous (L2-hot)
__global__ __launch_bounds__(256) void spmm_csr(const int* __restrict__ rowstart,
                                                const int* __restrict__ ecol,
                                                const float* __restrict__ eval,
                                                const float* __restrict__ x,
                                                const float* __restrict__ xprev,
                                                float* __restrict__ y,
                                                float alpha) {
    size_t tid = (size_t)blockIdx.x * 256 + threadIdx.x;   // V*64 = 6.4M
    int row = (int)(tid >> 6);
    int j = ((int)tid & 63) * 4;
    int s = rowstart[row];
    int e = rowstart[row + 1];
    float4 acc;
    if (xprev) {
        float4 p = *(const float4*)(xprev + (size_t)row * kC + j);
        acc = make_float4(-p.x, -p.y, -p.z, -p.w);
    } else {
        acc = make_float4(0.f, 0.f, 0.f, 0.f);
    }
    for (int i = s; i < e; ++i) {
        int c = ecol[i];
        float w = alpha * eval[i];
        float4 xv = *(const float4*)(x + (size_t)c * kC + j);
        acc.x += w * xv.x;
        acc.y += w * xv.y;
        acc.z += w * xv.z;
        acc.w += w * xv.w;
    }
    *(float4*)(y + (size_t)row * kC + j) = acc;
}

// ---------------- WMMA GEMM: out[b][v][o] = sum_kk As[v][kk] * Wt[o][kk] + bias[o]
__global__ __launch_bounds__(256) void cheb_gemm(const float* __restrict__ xbase,
                                                 const float* __restrict__ weight,
                                                 const float* __restrict__ bias,
                                                 float* __restrict__ out) {
    __shared__ float As[128 * LDA];     // 67.6 KB
    __shared__ float Wt[kFout * LDW];   // 33.8 KB

    const int t  = threadIdx.x;
    const int b  = blockIdx.y;
    const int v0 = blockIdx.x * 128;
    const size_t VC = (size_t)kV * kC;

    // Wt[o][kk] = weight[f][k][o], kk = k*32+f  (b128 reads over contiguous o)
#pragma unroll
    for (int i = 0; i < 8; ++i) {
        int idx4 = i * 256 + t;          // 2048 float4s = 8192 floats
        int kk = idx4 >> 4;              // 16 float4 (64 o) per kk
        int o  = (idx4 & 15) * 4;
        int k = kk >> 5, f = kk & 31;
        float4 w = *(const float4*)&weight[(size_t)f * (kK * kFout) + k * kFout + o];
        Wt[(o + 0) * LDW + kk] = w.x;
        Wt[(o + 1) * LDW + kk] = w.y;
        Wt[(o + 2) * LDW + kk] = w.z;
        Wt[(o + 3) * LDW + kk] = w.w;
    }
    // As[m][kk] = x_k[v0+m][b*32+f]  (b128 both sides; 8-lane groups cover 128B lines)
#pragma unroll
    for (int i = 0; i < 16; ++i) {
        int idx4 = i * 256 + t;          // 4096 float4s = 16384 floats
        int m = idx4 >> 5;               // 32 float4 (128 floats) per row
        int q = (idx4 & 31) * 4;         // kk, multiple of 4 (stays in one k-chunk)
        int v = v0 + m;
        float4 val = make_float4(0.f, 0.f, 0.f, 0.f);
        if (v < kV)
            val = *(const float4*)&xbase[(size_t)(q >> 5) * VC + (size_t)v * kC + b * kFin + (q & 31)];
        *(float4*)&As[m * LDA + q] = val;
    }
    __syncthreads();

    const int lane = t & 31;
    const int wave = t >> 5;
    const int l15  = lane & 15;
    const int h    = lane >> 4;          // fragment half: K = kk0 + 2h + j
    const int mrow = wave * 16 + l15;

    v8f acc[4] = {v8f{}, v8f{}, v8f{}, v8f{}};
    const float* ap  = &As[mrow * LDA + 2 * h];
    const float* wp0 = &Wt[(0 * 16 + l15) * LDW + 2 * h];
    const float* wp1 = &Wt[(1 * 16 + l15) * LDW + 2 * h];
    const float* wp2 = &Wt[(2 * 16 + l15) * LDW + 2 * h];
    const float* wp3 = &Wt[(3 * 16 + l15) * LDW + 2 * h];

#pragma unroll
    for (int kk0 = 0; kk0 < kKK; kk0 += 4) {
        v2f a  = *(const v2f*)(ap  + kk0);
        v2f b0 = *(const v2f*)(wp0 + kk0);
        v2f b1 = *(const v2f*)(wp1 + kk0);
        v2f b2 = *(const v2f*)(wp2 + kk0);
        v2f b3 = *(const v2f*)(wp3 + kk0);
        acc[0] = __builtin_amdgcn_wmma_f32_16x16x4_f32(false, a, false, b0, (short)0, acc[0], false, false);
        acc[1] = __builtin_amdgcn_wmma_f32_16x16x4_f32(false, a, false, b1, (short)0, acc[1], false, false);
        acc[2] = __builtin_amdgcn_wmma_f32_16x16x4_f32(false, a, false, b2, (short)0, acc[2], false, false);
        acc[3] = __builtin_amdgcn_wmma_f32_16x16x4_f32(false, a, false, b3, (short)0, acc[3], false, false);
    }

    // D layout: VGPR r -> row (wave*16 + r + 8*h), col (nt*16 + l15)
#pragma unroll
    for (int nt = 0; nt < 4; ++nt) {
        int n = nt * 16 + l15;
        float bv = bias[n];
#pragma unroll
        for (int r = 0; r < 8; ++r) {
            int m = wave * 16 + r + 8 * h;
            int v = v0 + m;
            if (v < kV)
                out[(size_t)b * kV * kFout + (size_t)v * kFout + n] = acc[nt][r] + bv;
        }
    }
}

extern "C" void kernel_launch(void* const* d_in, const int* in_sizes, int n_in,
                              void* d_out, int out_size, void* d_ws, size_t ws_size,
                              hipStream_t stream) {
    const float* inputs = (const float*)d_in[0];
    const int*   rows   = (const int*)d_in[1];
    const int*   cols   = (const int*)d_in[2];
    const float* vals   = (const float*)d_in[3];
    const float* weight = (const float*)d_in[4];
    const float* bias   = (const float*)d_in[5];
    float* out = (float*)d_out;

    const size_t N = (size_t)kV * kC;        // 25.6M floats per Chebyshev tensor
    float* x0 = (float*)d_ws;
    float* x1 = x0 + N;
    float* x2 = x1 + N;
    float* x3 = x2 + N;
    // CSR scratch after the Chebyshev tensors (~7 MB)
    int*   deg      = (int*)(x3 + N);
    int*   rowstart = deg + kV;          // kV+1 entries
    int*   pos      = rowstart + (kV + 1);
    int*   ecol     = pos + kV;
    float* eval     = (float*)(ecol + kE);

    const int tposeBlocks = (int)(N / 256);          // 100000
    const int vBlocks     = (kV + 255) / 256;        // 391
    const int eBlocks     = (kE + 255) / 256;        // 3125
    const int rowBlocks   = kV * 64 / 256;           // 25000

    // ---- build CSR (int atomics only) ----
    zero_ints<<<vBlocks, 256, 0, stream>>>(deg, kV);
    count_deg<<<eBlocks, 256, 0, stream>>>(rows, deg);
    scan_deg<<<1, 1024, 0, stream>>>(deg, rowstart);
    copy_ints<<<vBlocks, 256, 0, stream>>>(rowstart, pos, kV);
    fill_csr<<<eBlocks, 256, 0, stream>>>(rows, cols, vals, pos, ecol, eval);

    // ---- x0 = inputs^T  ([V][b*32+f]) ----
    transpose_in<<<tposeBlocks, 256, 0, stream>>>(inputs, x0);

    // ---- Chebyshev recurrence, gather-only SpMM ----
    // x1 = L x0
    spmm_csr<<<rowBlocks, 256, 0, stream>>>(rowstart, ecol, eval, x0, nullptr, x1, 1.0f);
    // x2 = 2 L x1 - x0
    spmm_csr<<<rowBlocks, 256, 0, stream>>>(rowstart, ecol, eval, x1, x0, x2, 2.0f);
    // x3 = 2 L x2 - x1
    spmm_csr<<<rowBlocks, 256, 0, stream>>>(rowstart, ecol, eval, x2, x1, x3, 2.0f);

    // ---- out = X @ Wflat + bias  via f32 WMMA ----
    dim3 grid((kV + 127) / 128, kB);
    cheb_gemm<<<grid, 256, 0, stream>>>(x0, weight, bias, out);
}